// CropImage_16527034155782
// MI455X (gfx1250) — compile-verified
//
#include <hip/hip_runtime.h>

// Crop + bilinear resize to 64x64, NHWC -> NCHW, f32.
// One block = one image x one 16-row output tile. 256 threads (8 wave32).

#define OUT 64
#define Y_OFF 82
#define IMG_H 644
#define IMG_W 640
#define ROWS_PER_BLOCK 16

__global__ __launch_bounds__(256) void crop_resize_kernel(
    const float* __restrict__ img,   // [B,644,640,3]
    const int*   __restrict__ bbox,  // [B,4] = x1,y1,x2,y2
    float*       __restrict__ out)   // [B,3,64,64]
{
    const int b  = blockIdx.x;
    const int ry = blockIdx.y;                 // output rows [ry*16, ry*16+16)
    const int t  = threadIdx.x;

    __shared__ int   sX0[OUT], sX1[OUT];
    __shared__ float sWX[OUT];
    __shared__ int   sY0[ROWS_PER_BLOCK], sY1[ROWS_PER_BLOCK];
    __shared__ float sWY[ROWS_PER_BLOCK];

    // Block-uniform bbox -> scalar loads.
    const int x1 = bbox[b * 4 + 0];
    const int y1 = bbox[b * 4 + 1] + Y_OFF;
    const int x2 = bbox[b * 4 + 2];
    const int y2 = bbox[b * 4 + 3] + Y_OFF;
    const int wi = x2 - x1;                    // crop width  (int)
    const int hi = y2 - y1;                    // crop height (int)
    const float wf = (float)wi;
    const float hf = (float)hi;

    if (t < OUT) {
        // x-axis sample table (all 64 columns)
        float sx = ((float)t + 0.5f) * wf * (1.0f / OUT) - 0.5f;
        sx = fminf(fmaxf(sx, 0.0f), wf - 1.0f);
        float x0f = floorf(sx);
        int   x0  = (int)x0f;
        int   xp  = min(x0 + 1, wi - 1);
        sX0[t] = x1 + x0;
        sX1[t] = x1 + xp;
        sWX[t] = sx - x0f;
    } else if (t < OUT + ROWS_PER_BLOCK) {
        // y-axis sample table (only this block's 16 rows)
        int   jl = t - OUT;
        int   j  = ry * ROWS_PER_BLOCK + jl;
        float sy = ((float)j + 0.5f) * hf * (1.0f / OUT) - 0.5f;
        sy = fminf(fmaxf(sy, 0.0f), hf - 1.0f);
        float y0f = floorf(sy);
        int   y0  = (int)y0f;
        int   yp  = min(y0 + 1, hi - 1);
        sY0[jl] = y1 + y0;
        sY1[jl] = y1 + yp;
        sWY[jl] = sy - y0f;
    }
    __syncthreads();

    const float* ibase = img + (size_t)b * ((size_t)IMG_H * IMG_W * 3);
    float*       obase = out + (size_t)b * (3 * OUT * OUT);

    // 16x64 tile = 1024 pixels, 256 threads -> 4 pixels each.
    // Lane-consecutive ox -> coalesced gathers along crop rows and
    // coalesced CHW stores.
#pragma unroll
    for (int k = 0; k < 4; ++k) {
        const int p   = t + (k << 8);          // 0..1023 in tile
        const int oyl = p >> 6;                // local row 0..15
        const int ox  = p & 63;

        const int   gy0 = sY0[oyl];
        const int   gy1 = sY1[oyl];
        const float wy  = sWY[oyl];
        const int   gx0 = sX0[ox];
        const int   gx1 = sX1[ox];
        const float wx  = sWX[ox];

        const float* r00 = ibase + (size_t)(gy0 * IMG_W + gx0) * 3;
        const float* r01 = ibase + (size_t)(gy0 * IMG_W + gx1) * 3;
        const float* r10 = ibase + (size_t)(gy1 * IMG_W + gx0) * 3;
        const float* r11 = ibase + (size_t)(gy1 * IMG_W + gx1) * 3;

        const float w00 = (1.0f - wx) * (1.0f - wy);
        const float w01 = wx * (1.0f - wy);
        const float w10 = (1.0f - wx) * wy;
        const float w11 = wx * wy;

        const int oy   = ry * ROWS_PER_BLOCK + oyl;
        const int opix = oy * OUT + ox;

#pragma unroll
        for (int c = 0; c < 3; ++c) {
            float v = r00[c] * w00 + r01[c] * w01
                    + r10[c] * w10 + r11[c] * w11;
            obase[c * (OUT * OUT) + opix] = v;
        }
    }
}

extern "C" void kernel_launch(void* const* d_in, const int* in_sizes, int n_in,
                              void* d_out, int out_size, void* d_ws, size_t ws_size,
                              hipStream_t stream) {
    (void)n_in; (void)out_size; (void)d_ws; (void)ws_size;
    const float* img  = (const float*)d_in[0];
    const int*   bbox = (const int*)d_in[1];
    float*       out  = (float*)d_out;

    const int B = in_sizes[1] / 4;             // bbox is [B,4]
    dim3 grid(B, OUT / ROWS_PER_BLOCK, 1);     // (64, 4)
    dim3 block(256, 1, 1);
    crop_resize_kernel<<<grid, block, 0, stream>>>(img, bbox, out);
}